// ContinuousDepthGene_25185688224004
// MI455X (gfx1250) — compile-verified
//
#include <hip/hip_runtime.h>
#include <hip/hip_bf16.h>

typedef __attribute__((ext_vector_type(2))) float v2f;
typedef __attribute__((ext_vector_type(8))) float v8f;

#define HID 128
#define LN_EPS 1e-5f

// ---------------------------------------------------------------------------
// WMMA GEMM: Z[nrows x 128] = H[nrows x K] @ W[K x 128] (+ bias)
// Block = 256 threads = 8 waves. Block tile = 64 rows x 128 cols.
// Wave w owns column tile w (16 cols) and iterates 4 row tiles (16 rows each).
// H tile staged in LDS (padded to kill bank conflicts); W read from L2.
// ---------------------------------------------------------------------------
template <int K, bool ADD_BIAS>
__global__ __launch_bounds__(256) void gemm_n128_wmma(
    const float* __restrict__ H, const float* __restrict__ W,
    const float* __restrict__ bias, float* __restrict__ Z, int nrows) {
  constexpr int LDK = K + 4;  // padded LDS row stride (floats)
  __shared__ float lds_h[64 * LDK];

  const int tid  = threadIdx.x;
  const int wave = tid >> 5;
  const int lane = tid & 31;
  const int block_row = blockIdx.x * 64;

  // Cooperative staged load of the 64 x K H-tile (float4, coalesced).
  const int nvec = (64 * K) / 4;
  for (int i = tid; i < nvec; i += 256) {
    const int r = (i * 4) / K;
    const int c = (i * 4) % K;
    const int gr = block_row + r;
    float4 v = make_float4(0.f, 0.f, 0.f, 0.f);
    if (gr < nrows) v = *(const float4*)(H + (size_t)gr * K + c);
    *(float4*)(&lds_h[r * LDK + c]) = v;
  }
  __syncthreads();

  const int c16   = wave * 16;       // this wave's column base
  const int m     = lane & 15;       // M (A) / N (B,C) index within tile
  const int khalf = (lane >> 4) * 2; // lanes 16-31 hold K+2,K+3 slices

  v8f acc[4];
#pragma unroll
  for (int r = 0; r < 4; ++r)
#pragma unroll
    for (int i = 0; i < 8; ++i) acc[r][i] = 0.f;

  for (int kt = 0; kt < K; kt += 4) {
    const int kb = kt + khalf;
    // B fragment (4x16 slice of W), reused across all 4 row tiles.
    v2f b;
    b.x = W[(size_t)(kb + 0) * HID + c16 + m];
    b.y = W[(size_t)(kb + 1) * HID + c16 + m];
#pragma unroll
    for (int r = 0; r < 4; ++r) {
      v2f a;
      a.x = lds_h[(r * 16 + m) * LDK + kb + 0];
      a.y = lds_h[(r * 16 + m) * LDK + kb + 1];
      acc[r] = __builtin_amdgcn_wmma_f32_16x16x4_f32(
          false, a, false, b, (short)0, acc[r], false, false);
    }
  }

  // C/D layout: VGPR i -> (M = i + 8*(lane>>4), N = lane&15).
#pragma unroll
  for (int r = 0; r < 4; ++r) {
    const int col = c16 + m;
#pragma unroll
    for (int i = 0; i < 8; ++i) {
      const int row = block_row + r * 16 + (lane >> 4) * 8 + i;
      if (row < nrows) {
        float v = acc[r][i];
        if (ADD_BIAS) v += bias[col];
        Z[(size_t)row * HID + col] = v;
      }
    }
  }
}

// ---------------------------------------------------------------------------
// Degree count (fp32 hardware atomics), then in-place dinv = rsqrt(deg + 1).
// ---------------------------------------------------------------------------
__global__ void deg_count(const long long* __restrict__ col, float* __restrict__ deg,
                          int nedges) {
  int e = blockIdx.x * blockDim.x + threadIdx.x;
  if (e < nedges) unsafeAtomicAdd(&deg[(int)col[e]], 1.0f);
}

__global__ void deg_to_dinv(float* __restrict__ d, int n) {
  int i = blockIdx.x * blockDim.x + threadIdx.x;
  if (i < n) d[i] = rsqrtf(d[i] + 1.0f);
}

// ---------------------------------------------------------------------------
// agg = dinv[i]^2 * z[i] + b_gcn   (self-loop term + bias), float4 vectorized.
// ---------------------------------------------------------------------------
__global__ void agg_init(const float* __restrict__ Z, const float* __restrict__ dinv,
                         const float* __restrict__ bias, float* __restrict__ Agg, int n) {
  int t = blockIdx.x * blockDim.x + threadIdx.x;  // one float4 per thread
  if (t >= n * 32) return;
  const int node = t >> 5;
  const int f4 = (t & 31) * 4;
  float d2 = dinv[node];
  d2 *= d2;
  const float4 z = *(const float4*)(Z + (size_t)node * HID + f4);
  const float4 b = *(const float4*)(bias + f4);
  float4 o;
  o.x = d2 * z.x + b.x;
  o.y = d2 * z.y + b.y;
  o.z = d2 * z.z + b.z;
  o.w = d2 * z.w + b.w;
  *(float4*)(Agg + (size_t)node * HID + f4) = o;
}

// ---------------------------------------------------------------------------
// Edge scatter: Agg[c] += dinv[r]*dinv[c] * Z[r].  One wave-lane handles a
// float4 chunk of features; 32 lanes cover 128 features of one edge.
// ---------------------------------------------------------------------------
__global__ void edge_scatter(const long long* __restrict__ row,
                             const long long* __restrict__ col,
                             const float* __restrict__ dinv,
                             const float* __restrict__ Z,
                             float* __restrict__ Agg, int nedges) {
  const long long tid = (long long)blockIdx.x * blockDim.x + threadIdx.x;
  const int e = (int)(tid >> 5);
  if (e >= nedges) return;
  const int lane = (int)(tid & 31);
  const int r = (int)row[e];
  const int c = (int)col[e];
  const float coef = dinv[r] * dinv[c];
  const float4 z = *(const float4*)(Z + (size_t)r * HID + lane * 4);
  float* dst = Agg + (size_t)c * HID + lane * 4;
  unsafeAtomicAdd(dst + 0, coef * z.x);
  unsafeAtomicAdd(dst + 1, coef * z.y);
  unsafeAtomicAdd(dst + 2, coef * z.z);
  unsafeAtomicAdd(dst + 3, coef * z.w);
}

// ---------------------------------------------------------------------------
// LayerNorm + tanh + Euler update, one wave32 per node.
// ---------------------------------------------------------------------------
__global__ void ln_tanh_update(const float* __restrict__ Agg,
                               const float* __restrict__ gamma,
                               const float* __restrict__ beta,
                               float* __restrict__ Hst, float dt, int n) {
  const int tid = blockIdx.x * blockDim.x + threadIdx.x;
  const int node = tid >> 5;
  if (node >= n) return;
  const int lane = tid & 31;

  float v[4];
  float s = 0.f;
#pragma unroll
  for (int j = 0; j < 4; ++j) {
    v[j] = Agg[(size_t)node * HID + j * 32 + lane];
    s += v[j];
  }
#pragma unroll
  for (int off = 16; off > 0; off >>= 1) s += __shfl_xor(s, off, 32);
  const float mu = s * (1.0f / HID);

  float vs = 0.f;
#pragma unroll
  for (int j = 0; j < 4; ++j) {
    const float d = v[j] - mu;
    vs += d * d;
  }
#pragma unroll
  for (int off = 16; off > 0; off >>= 1) vs += __shfl_xor(vs, off, 32);
  const float rstd = rsqrtf(vs * (1.0f / HID) + LN_EPS);

#pragma unroll
  for (int j = 0; j < 4; ++j) {
    const int f = j * 32 + lane;
    const float dh = tanhf((v[j] - mu) * rstd * gamma[f] + beta[f]);
    Hst[(size_t)node * HID + f] += dt * dh;
  }
}

// ---------------------------------------------------------------------------
// Column sums of H (for mean pooling), then out = mean(h) @ W_out + b_out.
// ---------------------------------------------------------------------------
__global__ void colsum_partial(const float* __restrict__ H, float* __restrict__ cs,
                               int n) {
  const int f = threadIdx.x;  // 128 threads
  const int start = blockIdx.x * 256;
  int end = start + 256;
  if (end > n) end = n;
  float s = 0.f;
  for (int r = start; r < end; ++r) s += H[(size_t)r * HID + f];
  unsafeAtomicAdd(&cs[f], s);
}

__global__ void out_proj(const float* __restrict__ cs, const float* __restrict__ Wout,
                         const float* __restrict__ bout, float* __restrict__ out,
                         float inv_n) {
  const int nf = threadIdx.x;  // 128 threads, 1 block
  float s = 0.f;
  for (int k = 0; k < HID; ++k) s += cs[k] * Wout[(size_t)k * HID + nf];
  out[nf] = s * inv_n + bout[nf];
}

// ---------------------------------------------------------------------------
extern "C" void kernel_launch(void* const* d_in, const int* in_sizes, int n_in,
                              void* d_out, int out_size, void* d_ws, size_t ws_size,
                              hipStream_t stream) {
  const float* x        = (const float*)d_in[0];
  const long long* ei   = (const long long*)d_in[1];  // int64 in reference
  const float* W_in     = (const float*)d_in[2];
  const float* b_in     = (const float*)d_in[3];
  const float* W_gcn    = (const float*)d_in[4];
  const float* b_gcn    = (const float*)d_in[5];
  const float* ln_gamma = (const float*)d_in[6];
  const float* ln_beta  = (const float*)d_in[7];
  const float* W_out    = (const float*)d_in[8];
  const float* b_out    = (const float*)d_in[9];
  float* out = (float*)d_out;

  const int N = in_sizes[0] / 64;   // 50000 nodes
  const int E = in_sizes[1] / 2;    // 600000 edges
  const long long* erow = ei;
  const long long* ecol = ei + E;

  // Workspace carve-up.
  uint8_t* ws = (uint8_t*)d_ws;
  const size_t hbytes = (size_t)N * HID * sizeof(float);
  float* h    = (float*)(ws);
  float* z    = (float*)(ws + hbytes);
  float* agg  = (float*)(ws + 2 * hbytes);
  float* dinv = (float*)(ws + 3 * hbytes);
  float* cs   = (float*)(ws + 3 * hbytes + ((size_t)N * sizeof(float) + 255 & ~(size_t)255));

  const float dt = 1.0f / 9.0f;  // DEPTH / (ODE_STEPS - 1)

  // 1) degrees -> dinv
  hipMemsetAsync(dinv, 0, (size_t)N * sizeof(float), stream);
  deg_count<<<(E + 255) / 256, 256, 0, stream>>>(ecol, dinv, E);
  deg_to_dinv<<<(N + 255) / 256, 256, 0, stream>>>(dinv, N);

  // 2) input projection: h = x @ W_in + b_in   (K = 64)
  gemm_n128_wmma<64, true><<<(N + 63) / 64, 256, 0, stream>>>(x, W_in, b_in, h, N);

  // 3) 9 Euler steps
  const int scatter_blocks = (int)(((long long)E * 32 + 255) / 256);
  for (int step = 0; step < 9; ++step) {
    gemm_n128_wmma<128, false><<<(N + 63) / 64, 256, 0, stream>>>(h, W_gcn, nullptr, z, N);
    agg_init<<<(N * 32 + 255) / 256, 256, 0, stream>>>(z, dinv, b_gcn, agg, N);
    edge_scatter<<<scatter_blocks, 256, 0, stream>>>(erow, ecol, dinv, z, agg, E);
    ln_tanh_update<<<(N * 32 + 255) / 256, 256, 0, stream>>>(agg, ln_gamma, ln_beta, h, dt, N);
  }

  // 4) mean pool + output projection (mean(h @ W + b) == mean(h) @ W + b)
  hipMemsetAsync(cs, 0, HID * sizeof(float), stream);
  colsum_partial<<<(N + 255) / 256, 128, 0, stream>>>(h, cs, N);
  out_proj<<<1, 128, 0, stream>>>(cs, W_out, b_out, out, 1.0f / (float)N);
}